// GMM_17978733101413
// MI455X (gfx1250) — compile-verified
//
#include <hip/hip_runtime.h>
#include <hip/hip_bf16.h>
#include <math.h>

// ---------------------------------------------------------------------------
// GMM EM iteration on gfx1250 (MI455X). FP32 WMMA (v_wmma_f32_16x16x4_f32)
// drives both big batched matmuls (whitening + covariance), ~34 GFLOP total.
// Operand matrices kept transposed in LDS with pitch 66 so every WMMA v2f
// fragment is a single aligned ds_load_b64 (no repack moves, no bank
// conflicts); E-step A fragments hoisted to registers across all 32 clusters.
// ---------------------------------------------------------------------------

typedef __attribute__((ext_vector_type(2))) float v2f;
typedef __attribute__((ext_vector_type(8))) float v8f;

#define N_PTS 65536
#define K_CL  32
#define D_DIM 64
#define GMM_EPS 1e-6f
#define LOG_2PI_F 1.8378770664093453f
#define CHUNK 4096   // rows per cov block
#define PITCH 66     // LDS row pitch: even (b64 aligned), 33 bank-pairs (conflict-free)

// workspace layout (in floats)
#define OFF_PREC    ((size_t)0)          // K*D*D = 131072
#define OFF_LOGDET  ((size_t)131072)     // K
#define OFF_MUP     ((size_t)131104)     // K*D = 2048
#define OFF_NK      ((size_t)133152)     // K
#define OFF_MEANNUM ((size_t)133184)     // K*D = 2048
#define OFF_SUMLL   ((size_t)135232)     // 1        (NK..SUMLL contiguous: 2081)
#define OFF_MEANSW  ((size_t)135240)     // K*D = 2048
#define OFF_NUM     ((size_t)137288)     // K*D*D = 131072
#define OFF_RESP    ((size_t)268360)     // N*K = 2097152

__device__ __forceinline__ v8f wmma_f32(v2f a, v2f b, v8f c) {
    // D(16x16,f32) = A(16x4,f32) * B(4x16,f32) + C
    return __builtin_amdgcn_wmma_f32_16x16x4_f32(
        /*neg_a=*/false, a, /*neg_b=*/false, b,
        /*c_mod=*/(short)0, c, /*reuse_a=*/false, /*reuse_b=*/false);
}

// ---------------------------------------------------------------------------
__global__ void zero_kernel(float* p, int n) {
    int i = blockIdx.x * blockDim.x + threadIdx.x;
    if (i < n) p[i] = 0.0f;
}

// ---------------------------------------------------------------------------
// Per-cluster Cholesky -> L, Linv = L^-1 (lower), prec = Linv^T,
// log_det = sum log diag(Linv), muP = mu @ prec.
// One block of 64 threads per cluster.
__global__ __launch_bounds__(64)
void chol_kernel(const float* __restrict__ sigma, const float* __restrict__ mu,
                 float* __restrict__ prec, float* __restrict__ logdet,
                 float* __restrict__ muP) {
    __shared__ float A[D_DIM][D_DIM + 1];
    __shared__ float Li[D_DIM][D_DIM + 1];
    const int tid = threadIdx.x;
    const int k = blockIdx.x;

    const float* S = sigma + (size_t)k * D_DIM * D_DIM;
    for (int j = 0; j < D_DIM; ++j) A[tid][j] = S[tid * D_DIM + j];
    __syncthreads();

    // in-place lower Cholesky
    for (int j = 0; j < D_DIM; ++j) {
        if (tid == j) {
            float s = A[j][j];
            for (int p = 0; p < j; ++p) s -= A[j][p] * A[j][p];
            A[j][j] = sqrtf(fmaxf(s, 1e-30f));
        }
        __syncthreads();
        if (tid > j) {
            float s = A[tid][j];
            for (int p = 0; p < j; ++p) s -= A[tid][p] * A[j][p];
            A[tid][j] = s / A[j][j];
        }
        __syncthreads();
    }

    // forward-substitute column tid of identity: Li = L^-1 (lower triangular)
    {
        const int c = tid;
        for (int i = 0; i < D_DIM; ++i) {
            if (i < c) { Li[i][c] = 0.0f; continue; }
            float s = (i == c) ? 1.0f : 0.0f;
            for (int p = c; p < i; ++p) s -= A[i][p] * Li[p][c];
            Li[i][c] = s / A[i][i];
        }
    }
    __syncthreads();

    // prec[d][e] = Li[e][d]
    float* P = prec + (size_t)k * D_DIM * D_DIM + tid * D_DIM;
    for (int e = 0; e < D_DIM; ++e) P[e] = Li[e][tid];

    // muP[e] = sum_d mu[d] * prec[d][e] = sum_d mu[d] * Li[e][d]
    {
        const float* muk = mu + k * D_DIM;
        float s = 0.0f;
        for (int d = 0; d < D_DIM; ++d) s += muk[d] * Li[tid][d];
        muP[k * D_DIM + tid] = s;
    }
    if (tid == 0) {
        float s = 0.0f;
        for (int i = 0; i < D_DIM; ++i) s += __logf(Li[i][i]);
        logdet[k] = s;
    }
}

// ---------------------------------------------------------------------------
// E-step: 64 rows of x per block, 8 waves. For each cluster: whitened tile
// via chained f32 WMMA (A fragments register-resident across clusters, B from
// transposed prec in LDS as aligned b64 pairs), Mahalanobis row-sums, then
// logsumexp/resp and atomic accumulation of Nk, sum(r*x), sum(lse).
__global__ __launch_bounds__(256)
void estep_kernel(const float* __restrict__ x, const float* __restrict__ pi,
                  const float* __restrict__ prec, const float* __restrict__ logdet,
                  const float* __restrict__ muP,
                  float* __restrict__ resp, float* __restrict__ NkG,
                  float* __restrict__ meansNumG, float* __restrict__ sumLLG) {
    __shared__ float xs[64][PITCH];     // x tile, row-major (n, d)
    __shared__ float psT[64][PITCH];    // prec_chol[k] transposed: psT[e][d]
    __shared__ float muPs[64];
    __shared__ float distLds[64][4];
    __shared__ float wlp[64][33];       // weighted log prob, later resp
    __shared__ float red[64];

    const int tid = threadIdx.x;
    const int lane = tid & 31;
    const int wave = tid >> 5;
    const int n0 = blockIdx.x * 64;

    {   // load x tile: thread -> (row, 16-col segment)
        int r = tid >> 2, c0 = (tid & 3) << 4;
        const float* src = x + (size_t)(n0 + r) * D_DIM + c0;
        #pragma unroll
        for (int j = 0; j < 16; ++j) xs[r][c0 + j] = src[j];
    }
    __syncthreads();

    // wave-static tile assignment: wave owns output rows m0..m0+15,
    // cols [e0,e0+16) and [e0+16,e0+32)
    const int tbase = wave * 2;
    const int m0 = (tbase >> 2) << 4;
    const int e0 = (tbase & 3) << 4;
    const int lm = lane & 15;
    const int koff = (lane < 16) ? 0 : 2;   // A/B K-split across half-waves
    const int arow = m0 + lm;

    // hoist A fragments (x tile) into registers: identical for every cluster
    v2f aReg[16];
    #pragma unroll
    for (int i = 0; i < 16; ++i)
        aReg[i] = *(const v2f*)&xs[arow][4 * i + koff];

    for (int k = 0; k < K_CL; ++k) {
        {   // load prec_chol[k] transposed + muP[k]
            int r = tid >> 2, c0 = (tid & 3) << 4;
            const float* src = prec + (size_t)k * 4096 + r * D_DIM + c0;
            #pragma unroll
            for (int j = 0; j < 16; ++j) psT[c0 + j][r] = src[j];
            if (tid < 64) muPs[tid] = muP[k * D_DIM + tid];
            if (k + 1 < K_CL)   // warm L0 for next cluster's prec
                __builtin_prefetch(prec + (size_t)(k + 1) * 4096 + r * D_DIM + c0, 0, 3);
        }
        __syncthreads();

        v8f c0v = {}, c1v = {};
        const float* bRow0 = &psT[e0 + lm][koff];
        const float* bRow1 = &psT[e0 + 16 + lm][koff];
        #pragma unroll
        for (int i = 0; i < 16; ++i) {
            v2f b0 = *(const v2f*)&bRow0[4 * i];
            v2f b1 = *(const v2f*)&bRow1[4 * i];
            c0v = wmma_f32(aReg[i], b0, c0v);
            c1v = wmma_f32(aReg[i], b1, c1v);
        }

        // (xP - muP)^2 row-sums; C layout: VGPR v -> rows m0+v / m0+v+8, N=lm
        float mp0 = muPs[e0 + lm];
        float mp1 = muPs[e0 + 16 + lm];
        #pragma unroll
        for (int v = 0; v < 8; ++v) {
            float y0 = c0v[v] - mp0, y1 = c1v[v] - mp1;
            float s0 = y0 * y0, s1 = y1 * y1;
            s0 += __shfl_xor(s0, 1, 16); s0 += __shfl_xor(s0, 2, 16);
            s0 += __shfl_xor(s0, 4, 16); s0 += __shfl_xor(s0, 8, 16);
            s1 += __shfl_xor(s1, 1, 16); s1 += __shfl_xor(s1, 2, 16);
            s1 += __shfl_xor(s1, 4, 16); s1 += __shfl_xor(s1, 8, 16);
            if ((lane & 15) == 0) {
                int rrow = m0 + v + ((lane >= 16) ? 8 : 0);
                distLds[rrow][(e0 >> 4) + 0] = s0;
                distLds[rrow][(e0 >> 4) + 1] = s1;
            }
        }
        __syncthreads();

        if (tid < 64) {
            float d = distLds[tid][0] + distLds[tid][1] +
                      distLds[tid][2] + distLds[tid][3];
            wlp[tid][k] = -0.5f * (64.0f * LOG_2PI_F + d) + logdet[k] + __logf(pi[k]);
        }
        __syncthreads();
    }

    // per-row logsumexp -> resp
    if (tid < 64) {
        float mx = -INFINITY;
        for (int k = 0; k < K_CL; ++k) mx = fmaxf(mx, wlp[tid][k]);
        float s = 0.0f;
        for (int k = 0; k < K_CL; ++k) s += __expf(wlp[tid][k] - mx);
        float lse = mx + __logf(s);
        red[tid] = lse;
        float* dst = resp + (size_t)(n0 + tid) * K_CL;
        for (int k = 0; k < K_CL; ++k) {
            float r = __expf(wlp[tid][k] - lse);
            wlp[tid][k] = r;     // reuse as resp tile
            dst[k] = r;
        }
    }
    __syncthreads();

    if (tid == 0) {              // log-likelihood partial
        float s = 0.0f;
        for (int i = 0; i < 64; ++i) s += red[i];
        unsafeAtomicAdd(sumLLG, s);
    }
    if (tid < K_CL) {            // Nk partial
        float s = 0.0f;
        for (int n = 0; n < 64; ++n) s += wlp[n][tid];
        unsafeAtomicAdd(&NkG[tid], s);
    }
    #pragma unroll               // sum_n resp[n][k]*x[n][d] partial (8 pairs/thread)
    for (int p = 0; p < 8; ++p) {
        int pair = tid * 8 + p;
        int k = pair >> 6, d = pair & 63;
        float s = 0.0f;
        for (int n = 0; n < 64; ++n) s += wlp[n][k] * xs[n][d];
        unsafeAtomicAdd(&meansNumG[k * D_DIM + d], s);
    }
}

// ---------------------------------------------------------------------------
// pi_new / means / LL
__global__ __launch_bounds__(256)
void small_kernel(const float* __restrict__ NkG, const float* __restrict__ meansNumG,
                  const float* __restrict__ sumLLG,
                  float* __restrict__ out, float* __restrict__ meansW) {
    __shared__ float pir[K_CL];
    __shared__ float tot;
    int tid = threadIdx.x;
    if (tid < K_CL) pir[tid] = NkG[tid] + GMM_EPS;
    __syncthreads();
    if (tid == 0) {
        float s = 0.0f;
        for (int i = 0; i < K_CL; ++i) s += pir[i];
        tot = s;
        out[0] = sumLLG[0] / (float)N_PTS;
    }
    __syncthreads();
    if (tid < K_CL) out[1 + tid] = pir[tid] / tot;
    for (int p = tid; p < K_CL * D_DIM; p += blockDim.x) {
        float m = meansNumG[p] / pir[p >> 6];
        out[33 + p] = m;
        meansW[p] = m;
    }
}

// ---------------------------------------------------------------------------
// Covariance numerator: num_k += (r.*(x-m_k))^T (x-m_k), all-WMMA reduction.
// Y stored transposed in LDS so A and B fragments (which pair along n) are
// aligned b64 loads. grid = (N/CHUNK, K), block = 256 (8 waves, 2 tiles each).
__global__ __launch_bounds__(256)
void cov_kernel(const float* __restrict__ x, const float* __restrict__ resp,
                const float* __restrict__ meansW, float* __restrict__ numG) {
    __shared__ float ysT[64][PITCH];   // (x - m_k) transposed: ysT[d][n]
    __shared__ float rs[64];
    __shared__ float ms[64];

    const int tid = threadIdx.x;
    const int lane = tid & 31;
    const int wave = tid >> 5;
    const int k = blockIdx.y;
    const int chunk0 = blockIdx.x * CHUNK;

    if (tid < 64) ms[tid] = meansW[k * D_DIM + tid];
    __syncthreads();

    const int tbase = wave * 2;
    const int d0 = (tbase >> 2) << 4;
    const int e0 = (tbase & 3) << 4;
    const int lm = lane & 15;
    const int koff = (lane < 16) ? 0 : 2;

    v8f acc0 = {}, acc1 = {};

    for (int nb = 0; nb < CHUNK; nb += 64) {
        const int n0 = chunk0 + nb;
        {   // load 64-row subtile, center by m_k, store transposed
            int r = tid >> 2, c0 = (tid & 3) << 4;
            const float* src = x + (size_t)(n0 + r) * D_DIM + c0;
            #pragma unroll
            for (int j = 0; j < 16; ++j) ysT[c0 + j][r] = src[j] - ms[c0 + j];
            if (tid < 64) rs[tid] = resp[(size_t)(n0 + tid) * K_CL + k];
            if (nb + 64 < CHUNK)   // stream next subtile toward L0/L2
                __builtin_prefetch(x + (size_t)(n0 + 64 + r) * D_DIM + c0, 0, 3);
        }
        __syncthreads();

        const float* aRow = &ysT[d0 + lm][koff];
        const float* bRow0 = &ysT[e0 + lm][koff];
        const float* bRow1 = &ysT[e0 + 16 + lm][koff];
        const float* rRow = &rs[koff];
        #pragma unroll
        for (int i = 0; i < 16; ++i) {
            v2f rv = *(const v2f*)&rRow[4 * i];
            v2f xa = *(const v2f*)&aRow[4 * i];
            v2f a;  a.x = rv.x * xa.x;  a.y = rv.y * xa.y;
            v2f b0 = *(const v2f*)&bRow0[4 * i];
            v2f b1 = *(const v2f*)&bRow1[4 * i];
            acc0 = wmma_f32(a, b0, acc0);
            acc1 = wmma_f32(a, b1, acc1);
        }
        __syncthreads();
    }

    // flush accumulators: VGPR v -> (d0+v / d0+v+8, e = e0(+16)+lm)
    #pragma unroll
    for (int v = 0; v < 8; ++v) {
        int d = d0 + v + ((lane >= 16) ? 8 : 0);
        unsafeAtomicAdd(&numG[((size_t)k * D_DIM + d) * D_DIM + e0 + lm], acc0[v]);
        unsafeAtomicAdd(&numG[((size_t)k * D_DIM + d) * D_DIM + e0 + 16 + lm], acc1[v]);
    }
}

// ---------------------------------------------------------------------------
__global__ void var_kernel(const float* __restrict__ numG,
                           const float* __restrict__ NkG,
                           float* __restrict__ out) {
    int idx = blockIdx.x * blockDim.x + threadIdx.x;
    if (idx >= K_CL * D_DIM * D_DIM) return;
    int k = idx >> 12;
    int de = idx & 4095;
    int d = de >> 6, e = de & 63;
    out[2081 + idx] = numG[idx] / NkG[k] + ((d == e) ? GMM_EPS : 0.0f);
}

// ---------------------------------------------------------------------------
extern "C" void kernel_launch(void* const* d_in, const int* in_sizes, int n_in,
                              void* d_out, int out_size, void* d_ws, size_t ws_size,
                              hipStream_t stream) {
    (void)in_sizes; (void)n_in; (void)out_size; (void)ws_size;
    const float* x     = (const float*)d_in[0];   // (N,D)
    const float* pi    = (const float*)d_in[1];   // (K,)
    const float* mu    = (const float*)d_in[2];   // (K,D)
    const float* sigma = (const float*)d_in[3];   // (K,D,D)
    float* out = (float*)d_out;
    float* ws  = (float*)d_ws;

    float* prec    = ws + OFF_PREC;
    float* logdet  = ws + OFF_LOGDET;
    float* muP     = ws + OFF_MUP;
    float* NkG     = ws + OFF_NK;
    float* meanNum = ws + OFF_MEANNUM;
    float* sumLL   = ws + OFF_SUMLL;
    float* meansW  = ws + OFF_MEANSW;
    float* numG    = ws + OFF_NUM;
    float* resp    = ws + OFF_RESP;

    zero_kernel<<<(2081 + 255) / 256, 256, 0, stream>>>(NkG, 2081);       // Nk+meansNum+sumLL
    zero_kernel<<<(131072 + 255) / 256, 256, 0, stream>>>(numG, 131072);  // cov numerator

    chol_kernel<<<K_CL, 64, 0, stream>>>(sigma, mu, prec, logdet, muP);

    estep_kernel<<<N_PTS / 64, 256, 0, stream>>>(x, pi, prec, logdet, muP,
                                                 resp, NkG, meanNum, sumLL);

    small_kernel<<<1, 256, 0, stream>>>(NkG, meanNum, sumLL, out, meansW);

    cov_kernel<<<dim3(N_PTS / CHUNK, K_CL), 256, 0, stream>>>(x, resp, meansW, numG);

    var_kernel<<<(K_CL * D_DIM * D_DIM + 255) / 256, 256, 0, stream>>>(numG, NkG, out);
}